// SSIMcal_51196010169096
// MI455X (gfx1250) — compile-verified
//
#include <hip/hip_runtime.h>
#include <stdint.h>

// SSIM, B=4 Z=32 H=W=512, 11x11 valid box filter -> out (B,1,1,1) = 4 floats.
// Memory-bound (268 MB in / 16 B out): stream rows HBM -> LDS via CDNA5 async
// copies (ASYNCcnt-tracked), sliding-window box sums, fused SSIM + reduction.

#define WIDTH  512
#define HEIGHT 512
#define OUTW   502
#define OUTH   502
#define ZDIM   32
#define BDIM   4
#define BAND   64
#define BANDS  8      // ceil(502/64)
#define RING   16     // LDS row ring (power of 2); window 11 + prefetch 4 < 16
#define PF     4      // rows prefetched ahead of the entering row

__device__ __forceinline__ uint32_t lds_addr_of(const void* p) {
    // Low 32 bits of a flat pointer into LDS == wave-relative LDS byte address.
    return (uint32_t)(uintptr_t)p;
}

__device__ __forceinline__ void async_load_f2(uint32_t lds_byte, const float* g) {
    // CDNA5 async global->LDS copy, 8B per lane, tracked with ASYNCcnt.
    asm volatile("global_load_async_to_lds_b64 %0, %1, off"
                 :: "v"(lds_byte), "v"((uint64_t)(uintptr_t)g)
                 : "memory");
}

__device__ __forceinline__ void wait_async_le8() {
    // Allow the newest PF(=4) rows x 2 instructions to stay in flight.
    asm volatile("s_wait_asynccnt 8" ::: "memory");
}

__device__ __forceinline__ float ssim_px(float sx, float sy, float sxx, float syy,
                                         float sxy, float C1, float C2) {
    const float inv = 1.0f / 121.0f;     // 1/NPIX
    const float cn  = 121.0f / 120.0f;   // COV_NORM
    float ux  = sx * inv,  uy  = sy * inv;
    float uxx = sxx * inv, uyy = syy * inv, uxy = sxy * inv;
    float vx  = cn * (uxx - ux * ux);
    float vy  = cn * (uyy - uy * uy);
    float vxy = cn * (uxy - ux * uy);
    float A1 = 2.0f * ux * uy + C1;
    float A2 = 2.0f * vxy + C2;
    float B1 = ux * ux + uy * uy + C1;
    float B2 = vx + vy + C2;
    return (A1 * A2) / (B1 * B2);
}

__global__ __launch_bounds__(256) void ssim_band_kernel(
        const float* __restrict__ img, const float* __restrict__ ref,
        const float* __restrict__ drange, float* __restrict__ accum) {
    __shared__ float histX[RING * WIDTH];   // 32 KB  img row ring
    __shared__ float histY[RING * WIDTH];   // 32 KB  ref row ring
    __shared__ float csum[5 * WIDTH];       // 10 KB  vertical column sums
    __shared__ float red[256];              //  1 KB  reduction

    const int tid  = threadIdx.x;
    const int col  = tid * 2;               // thread owns columns col, col+1
    const int blk  = blockIdx.x;
    const int iz   = blk / BANDS;           // image index 0..127
    const int band = blk % BANDS;
    const int b    = iz / ZDIM;             // batch index 0..3
    const int r0   = band * BAND;                   // first output row of band
    const int rows = min(BAND, OUTH - r0);          // 64 or 54
    const int inLast = r0 + rows - 1 + 10;          // last input row needed (<=511)

    const float dr = drange[b];
    const float C1 = (0.01f * dr) * (0.01f * dr);
    const float C2 = (0.03f * dr) * (0.03f * dr);

    const float* gi = img + (size_t)iz * (WIDTH * HEIGHT);
    const float* gr = ref + (size_t)iz * (WIDTH * HEIGHT);

    // ---- prologue: async-issue input rows r0 .. r0+13 into the ring --------
    for (int k = 0; k < 10 + PF; ++k) {
        int rr = r0 + k; if (rr > inLast) rr = inLast;
        const int slot = rr & (RING - 1);
        async_load_f2(lds_addr_of(&histX[slot * WIDTH + col]), gi + rr * WIDTH + col);
        async_load_f2(lds_addr_of(&histY[slot * WIDTH + col]), gr + rr * WIDTH + col);
    }
    wait_async_le8();   // rows r0..r0+9 resident in LDS

    // Per-thread vertical accumulators (2 columns x 5 channels).
    float sx0 = 0.f, sy0 = 0.f, sxx0 = 0.f, syy0 = 0.f, sxy0 = 0.f;
    float sx1 = 0.f, sy1 = 0.f, sxx1 = 0.f, syy1 = 0.f, sxy1 = 0.f;

    auto acc_slot = [&](int slot, float sgn) {
        const float x0 = histX[slot * WIDTH + col];
        const float x1 = histX[slot * WIDTH + col + 1];
        const float y0 = histY[slot * WIDTH + col];
        const float y1 = histY[slot * WIDTH + col + 1];
        sx0 += sgn * x0;       sy0 += sgn * y0;
        sxx0 += sgn * x0 * x0; syy0 += sgn * y0 * y0; sxy0 += sgn * x0 * y0;
        sx1 += sgn * x1;       sy1 += sgn * y1;
        sxx1 += sgn * x1 * x1; syy1 += sgn * y1 * y1; sxy1 += sgn * x1 * y1;
    };

    for (int k = 0; k < 10; ++k) acc_slot((r0 + k) & (RING - 1), 1.0f);

    float ssim_acc = 0.f;

    for (int r = r0; r < r0 + rows; ++r) {
        // Prefetch row r+14 (clamped so the wait-count stays uniform).
        {
            int rr = r + 10 + PF; if (rr > inLast) rr = inLast;
            const int slot = rr & (RING - 1);
            async_load_f2(lds_addr_of(&histX[slot * WIDTH + col]), gi + rr * WIDTH + col);
            async_load_f2(lds_addr_of(&histY[slot * WIDTH + col]), gr + rr * WIDTH + col);
        }
        wait_async_le8();                       // entering row r+10 resident
        acc_slot((r + 10) & (RING - 1), 1.0f);  // window now rows r..r+10

        csum[0 * WIDTH + col] = sx0;  csum[0 * WIDTH + col + 1] = sx1;
        csum[1 * WIDTH + col] = sy0;  csum[1 * WIDTH + col + 1] = sy1;
        csum[2 * WIDTH + col] = sxx0; csum[2 * WIDTH + col + 1] = sxx1;
        csum[3 * WIDTH + col] = syy0; csum[3 * WIDTH + col + 1] = syy1;
        csum[4 * WIDTH + col] = sxy0; csum[4 * WIDTH + col + 1] = sxy1;
        __syncthreads();

        if (tid <= 250) {   // output columns col, col+1 (502 outputs total)
            float t0[5], t1[5];
            #pragma unroll
            for (int ch = 0; ch < 5; ++ch) {
                const float* cp = &csum[ch * WIDTH + col];
                float s = cp[0];
                #pragma unroll
                for (int k = 1; k < 11; ++k) s += cp[k];
                t0[ch] = s;
                t1[ch] = s - cp[0] + cp[11];    // slide to next column
            }
            ssim_acc += ssim_px(t0[0], t0[1], t0[2], t0[3], t0[4], C1, C2);
            ssim_acc += ssim_px(t1[0], t1[1], t1[2], t1[3], t1[4], C1, C2);
        }
        __syncthreads();

        acc_slot(r & (RING - 1), -1.0f);        // row r leaves the window
    }

    // Workgroup reduction, then one atomic per block into accum[b].
    red[tid] = ssim_acc;
    __syncthreads();
    for (int s = 128; s > 0; s >>= 1) {
        if (tid < s) red[tid] += red[tid + s];
        __syncthreads();
    }
    if (tid == 0) atomicAdd(&accum[b], red[0]);
}

__global__ void ssim_zero_kernel(float* ws) {
    if (threadIdx.x < BDIM) ws[threadIdx.x] = 0.0f;
}

__global__ void ssim_final_kernel(const float* __restrict__ ws, float* __restrict__ out) {
    const int i = threadIdx.x;
    if (i < BDIM) out[i] = ws[i] * (1.0f / (float)(ZDIM * OUTH * OUTW));
}

extern "C" void kernel_launch(void* const* d_in, const int* in_sizes, int n_in,
                              void* d_out, int out_size, void* d_ws, size_t ws_size,
                              hipStream_t stream) {
    (void)in_sizes; (void)n_in; (void)out_size; (void)ws_size;
    const float* img = (const float*)d_in[0];
    const float* ref = (const float*)d_in[1];
    const float* dr  = (const float*)d_in[2];
    float* out = (float*)d_out;
    float* ws  = (float*)d_ws;

    ssim_zero_kernel<<<1, 32, 0, stream>>>(ws);
    ssim_band_kernel<<<BDIM * ZDIM * BANDS, 256, 0, stream>>>(img, ref, dr, ws);
    ssim_final_kernel<<<1, 32, 0, stream>>>(ws, out);
}